// AttentionPooling_46127948759432
// MI455X (gfx1250) — compile-verified
//
#include <hip/hip_runtime.h>
#include <hip/hip_bf16.h>
#include <math.h>

typedef __attribute__((ext_vector_type(2))) float v2f;
typedef __attribute__((ext_vector_type(8))) float v8f;

#define NN 1048576
#define CC 128
#define BB 8192

// Order-preserving float <-> uint mapping for atomicMax on scores.
__device__ __forceinline__ unsigned enc_f32(float f) {
    unsigned u = __float_as_uint(f);
    return u ^ (((unsigned)((int)u >> 31)) | 0x80000000u);
}
__device__ __forceinline__ float dec_f32(unsigned u) {
    unsigned b = (u & 0x80000000u) ? (u ^ 0x80000000u) : ~u;
    return __uint_as_float(b);
}

// ---------------- init: out = 0, segmax = 0 (== -inf sentinel), segsum = 0
__global__ void k_init(unsigned* __restrict__ segmax, float* __restrict__ segsum,
                       float* __restrict__ out) {
    int i = blockIdx.x * blockDim.x + threadIdx.x;
    if (i < BB) { segmax[i] = 0u; segsum[i] = 0.0f; }
    if (i < BB * CC) out[i] = 0.0f;
}

// ---------------- scores via V_WMMA_F32_16X16X4_F32 ----------------
// One wave handles 16 nodes. A = x tile [16 x 4] per step, B = query chunk
// replicated into all 16 columns, D accumulates over K=128 in 32 steps.
__global__ void k_scores(const float* __restrict__ x, const float* __restrict__ q,
                         const long long* __restrict__ batch,
                         float* __restrict__ scores, unsigned* __restrict__ segmax) {
    __shared__ float sq[CC];
    __shared__ float tileD[8][256];   // 8 waves/block, 8 VGPRs x 32 lanes each

    int t = threadIdx.x;
    if (t < CC) sq[t] = q[t];
    __syncthreads();

    int lane = t & 31;
    int wave = t >> 5;
    long long node0 = ((long long)blockIdx.x * 8 + wave) * 16;

    // A-layout (16x4 f32): lanes 0-15 -> rows 0-15 with K=0,1; lanes 16-31 -> K=2,3
    int m    = lane & 15;
    int koff = (lane >> 4) * 2;
    // B-layout (4x16 f32): VGPR0 lanes0-15=K0,lanes16-31=K1; VGPR1 lanes0-15=K2,lanes16-31=K3
    int bo0 = (lane < 16) ? 0 : 1;
    int bo1 = (lane < 16) ? 2 : 3;

    const float* arow = x + (node0 + m) * CC + koff;

    v8f acc = {};
#pragma unroll 8
    for (int j = 0; j < 32; ++j) {
        v2f a;
        a.x = arow[4 * j];
        a.y = arow[4 * j + 1];
        v2f b;
        b.x = sq[4 * j + bo0];
        b.y = sq[4 * j + bo1];
        acc = __builtin_amdgcn_wmma_f32_16x16x4_f32(
            /*neg_a=*/false, a, /*neg_b=*/false, b,
            /*c_mod=*/(short)0, acc, /*reuse_a=*/false, /*reuse_b=*/false);
    }

    // Spill D tile to LDS so lanes 0..15 can each grab their node's score.
#pragma unroll
    for (int v = 0; v < 8; ++v) tileD[wave][v * 32 + lane] = acc[v];
    __syncthreads();

    if (lane < 16) {
        long long n = node0 + lane;
        if (n < NN) {
            // node m score lives at (VGPR m, lanes 0-15) for m<8, (VGPR m-8, lanes 16-31) else
            float s = (lane < 8) ? tileD[wave][lane * 32]
                                 : tileD[wave][(lane - 8) * 32 + 16];
            scores[n] = s;
            int b = (int)batch[n];
            atomicMax(&segmax[b], enc_f32(s));
        }
    }
}

// ---------------- ex = exp(s - segmax), segsum += ex (in place) ----------------
__global__ void k_expsum(const long long* __restrict__ batch,
                         const unsigned* __restrict__ segmax,
                         float* __restrict__ ex, float* __restrict__ segsum) {
    int n = blockIdx.x * blockDim.x + threadIdx.x;
    if (n >= NN) return;
    int b = (int)batch[n];
    float e = __expf(ex[n] - dec_f32(segmax[b]));
    ex[n] = e;
    atomicAdd(&segsum[b], e);
}

// ---------------- reciprocal of segment sums ----------------
__global__ void k_rinv(const float* __restrict__ segsum, float* __restrict__ rinv) {
    int b = blockIdx.x * blockDim.x + threadIdx.x;
    if (b < BB) {
        float s = segsum[b];
        rinv[b] = (s > 0.0f) ? 1.0f / s : 0.0f;
    }
}

// ---------------- weighted pooling: out[b,:] += w[n] * x[n,:] ----------------
// 128 threads/block, one channel per thread, 256 contiguous (sorted) nodes per
// block; register accumulation, atomic flush only at segment boundaries.
__global__ void k_pool(const float* __restrict__ x, const long long* __restrict__ batch,
                       const float* __restrict__ ex, const float* __restrict__ rinv,
                       float* __restrict__ out) {
    __shared__ float wbuf[128];
    __shared__ int   bbuf[128];
    int t = threadIdx.x;
    int base0 = blockIdx.x * 256;

    float acc = 0.0f;
    int cur = -1;

    for (int base = base0; base < base0 + 256; base += 128) {
        int n = base + t;
        int b = (int)batch[n];
        wbuf[t] = ex[n] * rinv[b];
        bbuf[t] = b;
        __syncthreads();

        const float* xp = x + (size_t)base * CC + t;
        if (base + 128 < NN)
            __builtin_prefetch(xp + (size_t)128 * CC, 0, 0);   // global_prefetch_b8

        for (int i = 0; i < 128; ++i) {
            int b2 = bbuf[i];
            if (b2 != cur) {
                if (cur >= 0) atomicAdd(&out[(size_t)cur * CC + t], acc);
                acc = 0.0f;
                cur = b2;
            }
            acc = fmaf(wbuf[i], xp[(size_t)i * CC], acc);
        }
        __syncthreads();
    }
    if (cur >= 0) atomicAdd(&out[(size_t)cur * CC + t], acc);
}

extern "C" void kernel_launch(void* const* d_in, const int* in_sizes, int n_in,
                              void* d_out, int out_size, void* d_ws, size_t ws_size,
                              hipStream_t stream) {
    const float*     x     = (const float*)d_in[0];      // [N, C] f32
    const float*     q     = (const float*)d_in[1];      // [C] f32
    const long long* batch = (const long long*)d_in[2];  // [N] i64, sorted
    float*           out   = (float*)d_out;              // [B, C] f32

    char* ws = (char*)d_ws;
    float*    scores = (float*)ws;                                   // N floats (reused as ex)
    unsigned* segmax = (unsigned*)(ws + (size_t)NN * 4);             // B uints
    float*    segsum = (float*)(ws + (size_t)NN * 4 + (size_t)BB * 4);
    float*    rinv   = (float*)(ws + (size_t)NN * 4 + (size_t)BB * 8);

    k_init  <<<(BB * CC + 255) / 256, 256, 0, stream>>>(segmax, segsum, out);
    k_scores<<<NN / 16 / 8,           256, 0, stream>>>(x, q, batch, scores, segmax);
    k_expsum<<<(NN + 255) / 256,      256, 0, stream>>>(batch, segmax, scores, segsum);
    k_rinv  <<<(BB + 255) / 256,      256, 0, stream>>>(segsum, rinv);
    k_pool  <<<NN / 256,              128, 0, stream>>>(x, batch, scores, rinv, out);
}